// GAT_78176994721839
// MI455X (gfx1250) — compile-verified
//
#include <hip/hip_runtime.h>
#include <hip/hip_bf16.h>
#include <math.h>

// ---------------------------------------------------------------------------
// 2-layer GAT for gfx1250.  fp32 WMMA (V_WMMA_F32_16X16X4_F32) for the two
// dense projections; edge phases are bandwidth-bound scatter/gather with
// f32 global atomics.
// ---------------------------------------------------------------------------

typedef float v2f __attribute__((ext_vector_type(2)));
typedef float v8f __attribute__((ext_vector_type(8)));

// ------------------------- WMMA GEMM: Y = X @ W ----------------------------
// X: [nrows, 128] row-major, W: [128, NT*16] row-major, Y: [nrows, NT*16].
// 256 threads = 8 waves; wave w handles rows [blk*128 + w*16, +16).
// W is staged into LDS as K-pair-interleaved float2 so each lane's B fragment
// is one conflict-free ds_load_b64:  sB[kp*P + n] = { W[2kp][n], W[2kp+1][n] }
template <int NT>
__global__ void gemm_wmma_f32(const float* __restrict__ X,
                              const float* __restrict__ Wm,
                              float* __restrict__ Y,
                              int nrows)
{
    constexpr int NCOL = NT * 16;
    constexpr int P    = NCOL + 16;   // float2 pitch: +16 keeps the two 16-lane
                                      // halves on disjoint 32-bank halves
    __shared__ __align__(16) float2 sB[64 * P];

    const int tid = threadIdx.x;
    // cooperative interleaved stage of W into LDS
    for (int i = tid; i < 64 * NCOL; i += 256) {
        const int kp = i / NCOL;
        const int nn = i - kp * NCOL;
        float2 w2;
        w2.x = Wm[(2 * kp) * NCOL + nn];
        w2.y = Wm[(2 * kp + 1) * NCOL + nn];
        sB[kp * P + nn] = w2;
    }
    __syncthreads();

    const int wave = tid >> 5;
    const int lane = tid & 31;
    const int row0 = blockIdx.x * 128 + wave * 16;
    const int m    = lane & 15;
    const int khp  = lane >> 4;               // 0 for lanes 0-15, 1 for 16-31
    int row = row0 + m;
    if (row > nrows - 1) row = nrows - 1;     // clamp: garbage rows never stored

    const float2* Xp = (const float2*)(X + (size_t)row * 128) + khp;
    const int bbase = khp * P + m;            // float2 index into sB

    v8f acc[NT];
#pragma unroll
    for (int t = 0; t < NT; ++t) acc[t] = (v8f)0.0f;

    for (int kk = 0; kk < 32; ++kk) {         // K = 128 in steps of 4
        const float2 av = Xp[kk * 2];
        v2f a; a.x = av.x; a.y = av.y;
        const int bo = kk * 2 * P + bbase;
#pragma unroll
        for (int t = 0; t < NT; ++t) {
            const float2 bv = sB[bo + t * 16];
            v2f b; b.x = bv.x; b.y = bv.y;
            acc[t] = __builtin_amdgcn_wmma_f32_16x16x4_f32(
                false, a, false, b, (short)0, acc[t], false, false);
        }
    }

    // C/D layout: VGPR r holds row (r + (lane>=16 ? 8 : 0)), col = lane&15
    const int rbase = khp * 8;
#pragma unroll
    for (int r = 0; r < 8; ++r) {
        const int rr = row0 + rbase + r;
        if (rr < nrows) {
#pragma unroll
            for (int t = 0; t < NT; ++t)
                Y[(size_t)rr * NCOL + t * 16 + m] = acc[t][r];
        }
    }
}

// ------------------------- helpers -----------------------------------------
__device__ __forceinline__ void atomicMaxF32(float* addr, float v) {
    // bit-exact float max: int compare for >=0, uint-min for <0
    if (v >= 0.0f) atomicMax((int*)addr, __float_as_int(v));
    else           atomicMin((unsigned int*)addr, __float_as_uint(v));
}

__device__ __forceinline__ void edge_sd(const int* __restrict__ ei, int E,
                                        int e, int& s, int& d) {
    if (e < E) { s = ei[e]; d = ei[E + e]; }
    else       { s = e - E; d = e - E; }   // self loop
}

__device__ __forceinline__ float lrelu(float a) {
    return a > 0.0f ? a : 0.2f * a;
}

// ------------------------- layer-1 attention -------------------------------
// per (node, head): e_src/e_dst dot products + init amax/denom
__global__ void att_logits_l1(const float* __restrict__ xh,
                              const float* __restrict__ asrc,
                              const float* __restrict__ adst,
                              float* __restrict__ es, float* __restrict__ ed,
                              float* __restrict__ amax, float* __restrict__ dn,
                              int n)
{
    int t = blockIdx.x * blockDim.x + threadIdx.x;
    if (t >= n * 8) return;
    const int node = t >> 3, h = t & 7;
    const float* v  = xh + (size_t)node * 128 + h * 16;
    const float* as = asrc + h * 16;
    const float* ad = adst + h * 16;
    float s1 = 0.f, s2 = 0.f;
#pragma unroll
    for (int c = 0; c < 16; ++c) { float x = v[c]; s1 += x * as[c]; s2 += x * ad[c]; }
    es[t] = s1; ed[t] = s2;
    amax[t] = -__builtin_inff();
    dn[t]   = 0.0f;
}

__global__ void edge_max_l1(const int* __restrict__ ei, int E, int ET,
                            const float* __restrict__ es,
                            const float* __restrict__ ed,
                            float* __restrict__ amax)
{
    int t = blockIdx.x * blockDim.x + threadIdx.x;
    if (t >= ET * 8) return;
    const int e = t >> 3, h = t & 7;
    int s, d; edge_sd(ei, E, e, s, d);
    atomicMaxF32(&amax[d * 8 + h], lrelu(es[s * 8 + h] + ed[d * 8 + h]));
}

__global__ void edge_sum_l1(const int* __restrict__ ei, int E, int ET,
                            const float* __restrict__ es,
                            const float* __restrict__ ed,
                            const float* __restrict__ amax,
                            float* __restrict__ dn)
{
    int t = blockIdx.x * blockDim.x + threadIdx.x;
    if (t >= ET * 8) return;
    const int e = t >> 3, h = t & 7;
    int s, d; edge_sd(ei, E, e, s, d);
    const float a = lrelu(es[s * 8 + h] + ed[d * 8 + h]);
    atomicAdd(&dn[d * 8 + h], __expf(a - amax[d * 8 + h]));
}

__global__ void edge_scatter_l1(const int* __restrict__ ei, int E, int ET,
                                const float* __restrict__ es,
                                const float* __restrict__ ed,
                                const float* __restrict__ amax,
                                const float* __restrict__ dn,
                                const float* __restrict__ xh,
                                float* __restrict__ out)
{
    int t = blockIdx.x * blockDim.x + threadIdx.x;
    if (t >= ET * 8) return;
    const int e = t >> 3, h = t & 7;
    int s, d; edge_sd(ei, E, e, s, d);
    const float a   = lrelu(es[s * 8 + h] + ed[d * 8 + h]);
    const float att = __expf(a - amax[d * 8 + h]) / (dn[d * 8 + h] + 1e-16f);
    const float4* xs = (const float4*)(xh + (size_t)s * 128 + h * 16);
    float* o = out + (size_t)d * 128 + h * 16;
#pragma unroll
    for (int j = 0; j < 4; ++j) {
        const float4 v = xs[j];
        atomicAdd(o + j * 4 + 0, v.x * att);
        atomicAdd(o + j * 4 + 1, v.y * att);
        atomicAdd(o + j * 4 + 2, v.z * att);
        atomicAdd(o + j * 4 + 3, v.w * att);
    }
}

// bias + ELU in place (accumulator -> h1)
__global__ void finalize_l1(float* __restrict__ acc, const float* __restrict__ b,
                            int n)
{
    int t = blockIdx.x * blockDim.x + threadIdx.x;
    if (t >= n * 128) return;
    const float v = acc[t] + b[t & 127];
    acc[t] = v > 0.0f ? v : expm1f(v);
}

// pad W2 [128,40] -> [128,48] with zero cols
__global__ void pad_w2(const float* __restrict__ W2, float* __restrict__ W2p)
{
    int t = blockIdx.x * blockDim.x + threadIdx.x;
    if (t >= 128 * 48) return;
    const int k = t / 48, c = t % 48;
    W2p[t] = (c < 40) ? W2[k * 40 + c] : 0.0f;
}

// ------------------------- layer-2 attention (H=1, C=40) -------------------
__global__ void att_logits_l2(const float* __restrict__ xh2,
                              const float* __restrict__ asrc,
                              const float* __restrict__ adst,
                              float* __restrict__ es, float* __restrict__ ed,
                              float* __restrict__ amax, float* __restrict__ dn,
                              int n)
{
    int t = blockIdx.x * blockDim.x + threadIdx.x;
    if (t >= n) return;
    const float* v = xh2 + (size_t)t * 48;
    float s1 = 0.f, s2 = 0.f;
#pragma unroll
    for (int c = 0; c < 40; ++c) { float x = v[c]; s1 += x * asrc[c]; s2 += x * adst[c]; }
    es[t] = s1; ed[t] = s2;
    amax[t] = -__builtin_inff();
    dn[t]   = 0.0f;
}

__global__ void edge_max_l2(const int* __restrict__ ei, int E, int ET,
                            const float* __restrict__ es,
                            const float* __restrict__ ed,
                            float* __restrict__ amax)
{
    int t = blockIdx.x * blockDim.x + threadIdx.x;
    if (t >= ET) return;
    int s, d; edge_sd(ei, E, t, s, d);
    atomicMaxF32(&amax[d], lrelu(es[s] + ed[d]));
}

__global__ void edge_sum_l2(const int* __restrict__ ei, int E, int ET,
                            const float* __restrict__ es,
                            const float* __restrict__ ed,
                            const float* __restrict__ amax,
                            float* __restrict__ dn)
{
    int t = blockIdx.x * blockDim.x + threadIdx.x;
    if (t >= ET) return;
    int s, d; edge_sd(ei, E, t, s, d);
    const float a = lrelu(es[s] + ed[d]);
    atomicAdd(&dn[d], __expf(a - amax[d]));
}

__global__ void edge_scatter_l2(const int* __restrict__ ei, int E, int ET,
                                const float* __restrict__ es,
                                const float* __restrict__ ed,
                                const float* __restrict__ amax,
                                const float* __restrict__ dn,
                                const float* __restrict__ xh2,
                                float* __restrict__ out)
{
    int t = blockIdx.x * blockDim.x + threadIdx.x;
    if (t >= ET * 8) return;
    const int e = t >> 3, j = t & 7;           // 8 chunks of 5 channels
    int s, d; edge_sd(ei, E, e, s, d);
    const float a   = lrelu(es[s] + ed[d]);
    const float att = __expf(a - amax[d]) / (dn[d] + 1e-16f);
    const float* xs = xh2 + (size_t)s * 48 + j * 5;
    float* o = out + (size_t)d * 40 + j * 5;
#pragma unroll
    for (int c = 0; c < 5; ++c) atomicAdd(o + c, xs[c] * att);
}

__global__ void finalize_l2(float* __restrict__ out, const float* __restrict__ b,
                            int n)
{
    int t = blockIdx.x * blockDim.x + threadIdx.x;
    if (t >= n * 40) return;
    out[t] += b[t % 40];
}

// ---------------------------------------------------------------------------
extern "C" void kernel_launch(void* const* d_in, const int* in_sizes, int n_in,
                              void* d_out, int out_size, void* d_ws, size_t ws_size,
                              hipStream_t stream)
{
    (void)n_in; (void)out_size; (void)ws_size;

    const float* x   = (const float*)d_in[0];
    const int*   ei  = (const int*)d_in[1];
    const float* W1  = (const float*)d_in[2];
    const float* as1 = (const float*)d_in[3];
    const float* ad1 = (const float*)d_in[4];
    const float* b1  = (const float*)d_in[5];
    const float* W2  = (const float*)d_in[6];
    const float* as2 = (const float*)d_in[7];
    const float* ad2 = (const float*)d_in[8];
    const float* b2  = (const float*)d_in[9];
    float* out = (float*)d_out;

    const int n  = in_sizes[0] / 128;     // nodes
    const int E  = in_sizes[1] / 2;       // edges (before self-loops)
    const int ET = E + n;                 // with self-loops

    // workspace layout (256B aligned slices)
    char* ws = (char*)d_ws;
    size_t off = 0;
    auto alloc = [&](size_t bytes) -> float* {
        float* p = (float*)(ws + off);
        off += (bytes + 255) & ~(size_t)255;
        return p;
    };
    float* xh1  = alloc((size_t)n * 128 * 4);   // layer-1 proj; reused as xh2 [n,48]
    float* accB = alloc((size_t)n * 128 * 4);   // layer-1 accumulator -> h1
    float* es1  = alloc((size_t)n * 8 * 4);
    float* ed1  = alloc((size_t)n * 8 * 4);
    float* am1  = alloc((size_t)n * 8 * 4);
    float* dn1  = alloc((size_t)n * 8 * 4);
    float* es2  = alloc((size_t)n * 4);
    float* ed2  = alloc((size_t)n * 4);
    float* am2  = alloc((size_t)n * 4);
    float* dn2  = alloc((size_t)n * 4);
    float* W2p  = alloc((size_t)128 * 48 * 4);

    const int B = 256;
    const int eh = ET * 8;

    // zero accumulators (capture-safe memsets)
    hipMemsetAsync(accB, 0, (size_t)n * 128 * 4, stream);
    hipMemsetAsync(out, 0, (size_t)n * 40 * 4, stream);

    // ---- layer 1 ----
    gemm_wmma_f32<8><<<(n + 127) / 128, 256, 0, stream>>>(x, W1, xh1, n);
    att_logits_l1<<<(n * 8 + B - 1) / B, B, 0, stream>>>(xh1, as1, ad1, es1, ed1, am1, dn1, n);
    edge_max_l1<<<(eh + B - 1) / B, B, 0, stream>>>(ei, E, ET, es1, ed1, am1);
    edge_sum_l1<<<(eh + B - 1) / B, B, 0, stream>>>(ei, E, ET, es1, ed1, am1, dn1);
    edge_scatter_l1<<<(eh + B - 1) / B, B, 0, stream>>>(ei, E, ET, es1, ed1, am1, dn1, xh1, accB);
    finalize_l1<<<(n * 128 + B - 1) / B, B, 0, stream>>>(accB, b1, n);

    // ---- layer 2 ----
    pad_w2<<<(128 * 48 + B - 1) / B, B, 0, stream>>>(W2, W2p);
    float* xh2 = xh1;   // xh1 dead after edge_scatter_l1
    gemm_wmma_f32<3><<<(n + 127) / 128, 256, 0, stream>>>(accB, W2p, xh2, n);
    att_logits_l2<<<(n + B - 1) / B, B, 0, stream>>>(xh2, as2, ad2, es2, ed2, am2, dn2, n);
    edge_max_l2<<<(ET + B - 1) / B, B, 0, stream>>>(ei, E, ET, es2, ed2, am2);
    edge_sum_l2<<<(ET + B - 1) / B, B, 0, stream>>>(ei, E, ET, es2, ed2, am2, dn2);
    edge_scatter_l2<<<(eh + B - 1) / B, B, 0, stream>>>(ei, E, ET, es2, ed2, am2, dn2, xh2, out);
    finalize_l2<<<(n * 40 + B - 1) / B, B, 0, stream>>>(out, b2, n);
}